// ScoreBranch_89687507075235
// MI455X (gfx1250) — compile-verified
//
#include <hip/hip_runtime.h>

// Problem shape (fixed by the reference).
#define B_DIM 16
#define T_DIM 2048
#define BLOCK 256
#define CPT   (T_DIM / BLOCK)   // 8 contiguous columns per thread

typedef __attribute__((ext_vector_type(4))) float v4f;  // native vector: OK for nontemporal builtins

__device__ __forceinline__ float fast_sigmoid(float x) {
    const float L2E = 1.44269504088896340736f;
    // 1 / (1 + 2^(-x*log2(e)))  ->  v_exp_f32 + v_rcp_f32
    return __builtin_amdgcn_rcpf(1.0f + __builtin_amdgcn_exp2f(-x * L2E));
}

__global__ void sigmoid_kernel(const float* __restrict__ logits,
                               float* __restrict__ scores, int n) {
    int i = blockIdx.x * blockDim.x + threadIdx.x;
    if (i < n) scores[i] = fast_sigmoid(logits[i]);
}

__global__ void __launch_bounds__(BLOCK)
row_softmax_kernel(const float* __restrict__ scores,
                   const int* __restrict__ seq_len,
                   float* __restrict__ out) {
    // Single shared allocation -> LDS offset 0 for the async-copy destination.
    // [0, T_DIM): column scores   [T_DIM, T_DIM+8): per-wave partial sums
    __shared__ __align__(16) float smem[T_DIM + 8];

    const int row = blockIdx.x;          // row in [0, B*T)
    const int b   = row >> 11;           // / T_DIM
    const int i   = row & (T_DIM - 1);   // % T_DIM
    const int L   = seq_len[b];
    const int tid = threadIdx.x;
    float* outRow = out + (size_t)row * T_DIM;
    v4f* o4       = reinterpret_cast<v4f*>(outRow) + tid * 2;

    if (i >= L) {   // invalid row: stream zeros, done (uniform branch)
        v4f z = (v4f){0.f, 0.f, 0.f, 0.f};
        __builtin_nontemporal_store(z, o4);
        __builtin_nontemporal_store(z, o4 + 1);
        return;
    }

    // ---- CDNA5 async copy: stage the 8KB score row into LDS ----
    // 32 bytes per lane = 2 x b128; INST_OFFSET applies to both LDS and
    // global addresses, so one VGPR pair + one LDS-addr VGPR serve both.
    {
        unsigned lds_byte = (unsigned)(tid * 32u);                    // smem @ LDS offset 0
        unsigned long long ga =
            (unsigned long long)(const void*)(scores + (size_t)b * T_DIM) +
            (unsigned long long)(tid * 32u);
        asm volatile("global_load_async_to_lds_b128 %0, %1, off"
                     :: "v"(lds_byte), "v"(ga) : "memory");
        asm volatile("global_load_async_to_lds_b128 %0, %1, off offset:16"
                     :: "v"(lds_byte), "v"(ga) : "memory");
        asm volatile("s_wait_asynccnt 0" ::: "memory");
    }
    __syncthreads();

    const float si = smem[i];            // LDS broadcast read

    // This thread's 8 column scores: 2 x ds_load_b128, conflict-free.
    v4f c0 = reinterpret_cast<const v4f*>(smem)[tid * 2];
    v4f c1 = reinterpret_cast<const v4f*>(smem)[tid * 2 + 1];
    float sj[CPT] = {c0.x, c0.y, c0.z, c0.w, c1.x, c1.y, c1.z, c1.w};

    const float L2E = 1.44269504088896340736f;
    float e[CPT];
    float local = 0.f;
    const int j0 = tid * CPT;
#pragma unroll
    for (int k = 0; k < CPT; ++k) {
        float d  = fabsf(si - sj[k]);
        float t  = (0.5f - d) * 10.0f;                 // (1 - |d| - 0.5)/0.1
        float r  = fast_sigmoid(t);                    // rel in (0,1)
        float ev = __builtin_amdgcn_exp2f(r * L2E);    // exp(rel), no max needed
        ev = (j0 + k < L) ? ev : 0.0f;                 // masked col -> exact 0
        e[k] = ev;
        local += ev;
    }

    // Deterministic block reduction: wave32 xor-shuffle tree + LDS partials.
#pragma unroll
    for (int m = 16; m > 0; m >>= 1) local += __shfl_xor(local, m, 32);
    const int lane = tid & 31, wid = tid >> 5;
    if (lane == 0) smem[T_DIM + wid] = local;
    __syncthreads();
    float total = 0.f;
#pragma unroll
    for (int w = 0; w < BLOCK / 32; ++w) total += smem[T_DIM + w];
    const float inv = __builtin_amdgcn_rcpf(total);    // L>=1 => total>0

    // Pass 2: scale registers, stream out with non-temporal b128 stores.
    v4f r0 = (v4f){e[0]*inv, e[1]*inv, e[2]*inv, e[3]*inv};
    v4f r1 = (v4f){e[4]*inv, e[5]*inv, e[6]*inv, e[7]*inv};
    __builtin_nontemporal_store(r0, o4);
    __builtin_nontemporal_store(r1, o4 + 1);
}

extern "C" void kernel_launch(void* const* d_in, const int* in_sizes, int n_in,
                              void* d_out, int out_size, void* d_ws, size_t ws_size,
                              hipStream_t stream) {
    const float* logits  = (const float*)d_in[0];   // [B, T, 1] fp32
    const int*   seqlen  = (const int*)d_in[1];     // [B] int32
    float*       out     = (float*)d_out;           // [B, T, T] fp32
    float*       scores  = (float*)d_ws;            // B*T fp32 scratch (128 KB)

    const int n = in_sizes[0];                      // B*T
    sigmoid_kernel<<<(n + 255) / 256, 256, 0, stream>>>(logits, scores, n);
    row_softmax_kernel<<<n, BLOCK, 0, stream>>>(scores, seqlen, out);
}